// EdgeNetwork_41832981463620
// MI455X (gfx1250) — compile-verified
//
#include <hip/hip_runtime.h>

typedef __attribute__((ext_vector_type(2))) float v2f;
typedef __attribute__((ext_vector_type(8))) float v8f;

#define B_ 8
#define N_ 2048
#define E_ 4096
#define D_ 3
#define H_ 8

#define WAVES   8
#define TILE_E  16
#define BLOCK_E (WAVES * TILE_E)   // 128 edges per workgroup

// LDS: Xpad[N][4] (32KB, col3 = 0) + epilogue staging 8 waves * 16 e * 8 f32 (4KB)
__global__ __launch_bounds__(256, 1)
void edge_network_kernel(const float* __restrict__ X,
                         const float* __restrict__ Ri,
                         const float* __restrict__ Ro,
                         const float* __restrict__ W1,
                         const float* __restrict__ b1,
                         const float* __restrict__ W2,
                         const float* __restrict__ b2,
                         float* __restrict__ out)
{
    __shared__ float Xs[N_ * 4 + WAVES * TILE_E * 8];

    const int b    = blockIdx.y;
    const int tid  = threadIdx.x;
    const int lane = tid & 31;
    const int wave = tid >> 5;
    const int col  = lane & 15;   // N-index of B/C/D, M-index of A
    const int half = lane >> 4;   // lane half selects K pairs

    // ---- stage X[b] into LDS, padded to 4 cols (col 3 == 0) ----
    const float* Xb = X + (size_t)b * N_ * D_;
    for (int w = tid; w < N_ * 4; w += 256) {
        const int n = w >> 2, d = w & 3;
        Xs[w] = (d < 3) ? Xb[n * 3 + d] : 0.0f;
    }
    __syncthreads();

    const int e0 = blockIdx.x * BLOCK_E + wave * TILE_E;

    // A-matrix (f32 16x4) per ISA layout: VGPR j, lane L -> M = L&15, K = 2*(L>>4)+j
    // address: Ro[b, n0 + K, e0 + M]; e contiguous -> 64B coalesced per half-wave
    const size_t based = (size_t)b * N_ * E_ + (size_t)(2 * half) * E_ + (size_t)(e0 + col);
    const float* pro = Ro + based;
    const float* pri = Ri + based;

    v8f acc_o = {};   // bo tile: rows M = e, cols 0..2 = d
    v8f acc_i = {};   // bi tile

    for (int n0 = 0; n0 < N_; n0 += 16) {
#pragma unroll
        for (int u = 0; u < 4; ++u) {
            const int    nn  = n0 + u * 4;
            const size_t row = (size_t)nn * E_;
            // Ri/Ro are single-use streams: non-temporal loads keep them out of L2
            v2f a_o, a_i;
            a_o.x = __builtin_nontemporal_load(pro + row);        // K = 2*half
            a_o.y = __builtin_nontemporal_load(pro + row + E_);   // K = 2*half + 1
            a_i.x = __builtin_nontemporal_load(pri + row);
            a_i.y = __builtin_nontemporal_load(pri + row + E_);
            // B-matrix (f32 4x16): VGPR j, lane L -> K = 2*(L>>4)+j, N = L&15
            const int kb = nn + 2 * half;
            const float t0 = Xs[kb * 4 + (col & 3)];
            const float t1 = Xs[(kb + 1) * 4 + (col & 3)];
            v2f bm;
            bm.x = (col < 4) ? t0 : 0.0f;   // cols 3..15 contribute zero
            bm.y = (col < 4) ? t1 : 0.0f;
            acc_o = __builtin_amdgcn_wmma_f32_16x16x4_f32(false, a_o, false, bm,
                                                          (short)0, acc_o, false, false);
            acc_i = __builtin_amdgcn_wmma_f32_16x16x4_f32(false, a_i, false, bm,
                                                          (short)0, acc_i, false, false);
        }
    }

    // ---- stage accumulators through LDS: per-wave [16 e][8]: 0-2 bo, 3-5 bi ----
    // C/D layout: VGPR v, lane L -> M = v + 8*(L>>4), N = L&15
    float* Es = Xs + N_ * 4 + wave * TILE_E * 8;
    if (col < 3) {
#pragma unroll
        for (int v = 0; v < 8; ++v) {
            const int el = v + 8 * half;
            Es[el * 8 + col]     = acc_o[v];
            Es[el * 8 + 3 + col] = acc_i[v];
        }
    }
    __syncthreads();

    // ---- fused per-edge MLP: h = tanh(W1 @ [bo;bi] + b1); out = sigmoid(W2 @ h + b2) ----
    if (lane < 16) {
        float f[6];
#pragma unroll
        for (int c = 0; c < 6; ++c) f[c] = Es[lane * 8 + c];
        float s = 0.0f;
#pragma unroll
        for (int j = 0; j < H_; ++j) {
            float a = b1[j];
#pragma unroll
            for (int c = 0; c < 6; ++c) a = fmaf(W1[j * 6 + c], f[c], a);
            s = fmaf(W2[j], tanhf(a), s);
        }
        s += b2[0];
        const float r = 1.0f / (1.0f + __expf(-s));
        __builtin_nontemporal_store(r, out + (size_t)b * E_ + e0 + lane);
    }
}

extern "C" void kernel_launch(void* const* d_in, const int* in_sizes, int n_in,
                              void* d_out, int out_size, void* d_ws, size_t ws_size,
                              hipStream_t stream) {
    const float* X  = (const float*)d_in[0];
    const float* Ri = (const float*)d_in[1];
    const float* Ro = (const float*)d_in[2];
    const float* W1 = (const float*)d_in[3];
    const float* b1 = (const float*)d_in[4];
    const float* W2 = (const float*)d_in[5];
    const float* b2 = (const float*)d_in[6];
    float* out = (float*)d_out;

    dim3 grid(E_ / BLOCK_E, B_);   // 32 x 8 = 256 workgroups, 8 waves each
    edge_network_kernel<<<grid, dim3(256), 0, stream>>>(X, Ri, Ro, W1, b1, W2, b2, out);
}